// MessagePassingConvolution_48395691492084
// MI455X (gfx1250) — compile-verified
//
#include <hip/hip_runtime.h>
#include <math.h>

// CDNA5 / gfx1250, wave32. f32 WMMA for the per-edge MLP GEMMs.
typedef float v2f __attribute__((ext_vector_type(2)));
typedef float v8f __attribute__((ext_vector_type(8)));

#define MUL 32
#define F_HID 64          // hidden width of the edge MLP
#define F_MIX 128         // mix width (4*MUL)
#define EDGES_PER_WAVE 16 // M dimension of WMMA tiles
#define WAVES_PER_BLOCK 8
#define EDGES_PER_BLOCK (EDGES_PER_WAVE * WAVES_PER_BLOCK)
#define INV_SQRT3 0.57735026918962576451f
#define MSG_SCALE 0.25f   // 1/sqrt(AVG_NUM_NEIGHBORS=16)

__device__ __forceinline__ float swishf(float x) {
    return x / (1.0f + __expf(-x));
}

__global__ __launch_bounds__(256)
void zero_out_kernel(float* __restrict__ out, long long n) {
    long long i = (long long)blockIdx.x * blockDim.x + threadIdx.x;
    long long stride = (long long)gridDim.x * blockDim.x;
    for (; i < n; i += stride) out[i] = 0.0f;
}

__global__ __launch_bounds__(256)
void mpconv_kernel(const float* __restrict__ node_feats,  // [N,128]
                   const float* __restrict__ edge_attrs,  // [E,4]
                   const int*   __restrict__ senders,     // [E]
                   const int*   __restrict__ receivers,   // [E]
                   const float* __restrict__ w0,          // [64]
                   const float* __restrict__ w1,          // [64,64]
                   const float* __restrict__ w2,          // [64,128]
                   float*       __restrict__ out,         // [N,256]
                   int n_edges)
{
    __shared__ float sH[EDGES_PER_BLOCK * F_HID];    // 32 KB: H0 then H1 per wave-tile
    __shared__ float sMIX[EDGES_PER_BLOCK * F_MIX];  // 64 KB: per-edge mix vector

    const int tid  = threadIdx.x;
    const int lane = tid & 31;
    const int wave = tid >> 5;
    const int base = blockIdx.x * EDGES_PER_BLOCK;

    // ---------------- Phase A: H0 = swish(ea0 * w0), block-wide ----------------
    for (int idx = tid; idx < EDGES_PER_BLOCK * F_HID; idx += 256) {
        int m = idx >> 6;        // local edge
        int j = idx & 63;        // feature
        int e = base + m; if (e >= n_edges) e = n_edges - 1;
        float x = edge_attrs[(long long)e * 4] * w0[j];
        sH[idx] = swishf(x);
    }
    __syncthreads();

    // WMMA f32 16x16x4 A-fragment lane mapping (32-bit A 16x4 layout):
    // lanes 0-15 hold rows M=lane with K = k,k+1 ; lanes 16-31 hold K = k+2,k+3
    const int halfsel = lane >> 4;                    // 0 or 1
    const int mrow    = lane & 15;                    // matrix row / B column
    const int rowBase = (wave * EDGES_PER_WAVE + mrow) * F_HID;

    v2f aF[16];
    #pragma unroll
    for (int t = 0; t < 16; ++t) {
        int k = 4 * t + halfsel * 2;
        aF[t] = *(const v2f*)&sH[rowBase + k];        // ds_load_b64
    }

    // ---------------- Phase B: H1 = swish((H0 @ W1) / 8) ----------------
    #pragma unroll
    for (int nt = 0; nt < 4; ++nt) {                  // 4 N-tiles of 16
        v8f acc = {};
        const int col = nt * 16 + mrow;
        #pragma unroll
        for (int t = 0; t < 16; ++t) {                // K = 64 in steps of 4
            int row = 4 * t + halfsel * 2;
            v2f b;                                    // B 4x16: row striped over lanes
            b.x = w1[row * 64 + col];
            b.y = w1[(row + 1) * 64 + col];
            acc = __builtin_amdgcn_wmma_f32_16x16x4_f32(
                      false, aF[t], false, b, (short)0, acc, false, false);
        }
        // D layout: reg r holds row (r + halfsel*8), column = col
        #pragma unroll
        for (int r = 0; r < 8; ++r) {
            int mr = r + halfsel * 8;
            sH[(wave * EDGES_PER_WAVE + mr) * F_HID + col] = swishf(acc[r] * 0.125f);
        }
    }
    __syncthreads();

    // ---------------- Phase C: mix = (H1 @ W2) / 8 ----------------
    #pragma unroll
    for (int t = 0; t < 16; ++t) {
        int k = 4 * t + halfsel * 2;
        aF[t] = *(const v2f*)&sH[rowBase + k];
    }
    #pragma unroll
    for (int nt = 0; nt < 8; ++nt) {                  // 8 N-tiles of 16 (N=128)
        v8f acc = {};
        const int col = nt * 16 + mrow;
        #pragma unroll
        for (int t = 0; t < 16; ++t) {
            int row = 4 * t + halfsel * 2;
            v2f b;
            b.x = w2[row * 128 + col];
            b.y = w2[(row + 1) * 128 + col];
            acc = __builtin_amdgcn_wmma_f32_16x16x4_f32(
                      false, aF[t], false, b, (short)0, acc, false, false);
        }
        #pragma unroll
        for (int r = 0; r < 8; ++r) {
            int mr = r + halfsel * 8;
            sMIX[(wave * EDGES_PER_WAVE + mr) * F_MIX + col] = acc[r] * 0.125f;
        }
    }
    __syncthreads();

    // ---------------- Phase D: tensor-product messages + scatter-add ----------------
    // One wave per 16 edges; 32 lanes = 32 channels (MUL).
    const int c = lane;
    for (int m = 0; m < EDGES_PER_WAVE; ++m) {
        int e = base + wave * EDGES_PER_WAVE + m;
        if (e >= n_edges) break;
        int snd = senders[e];
        int rcv = receivers[e];
        float ex = edge_attrs[(long long)e * 4 + 1];
        float ey = edge_attrs[(long long)e * 4 + 2];
        float ez = edge_attrs[(long long)e * 4 + 3];

        const float* nf = node_feats + (long long)snd * (4 * MUL);
        float s  = nf[c];                    // scalar channel c
        float v0 = nf[32 + 3 * c + 0];       // vector channel c (x,y,z)
        float v1 = nf[32 + 3 * c + 1];
        float v2 = nf[32 + 3 * c + 2];

        const float* mixrow = &sMIX[(wave * EDGES_PER_WAVE + m) * F_MIX];
        float mix0 = mixrow[c];              // scales s_e
        float mix1 = mixrow[32 + c];         // scales tp_0e
        float mix2 = mixrow[64 + c];         // scales v_e
        float mix3 = mixrow[96 + c];         // scales tp_1o

        float tp0e = (v0 * ex + v1 * ey + v2 * ez) * INV_SQRT3;

        float* ob = out + (long long)rcv * 256;
        unsafeAtomicAdd(&ob[c],        s    * mix0 * MSG_SCALE);
        unsafeAtomicAdd(&ob[32 + c],   tp0e * mix1 * MSG_SCALE);
        float vm = mix2 * MSG_SCALE;
        unsafeAtomicAdd(&ob[64 + 3 * c + 0], v0 * vm);
        unsafeAtomicAdd(&ob[64 + 3 * c + 1], v1 * vm);
        unsafeAtomicAdd(&ob[64 + 3 * c + 2], v2 * vm);
        float tm = s * mix3 * MSG_SCALE;
        unsafeAtomicAdd(&ob[160 + 3 * c + 0], tm * ex);
        unsafeAtomicAdd(&ob[160 + 3 * c + 1], tm * ey);
        unsafeAtomicAdd(&ob[160 + 3 * c + 2], tm * ez);
    }
}

extern "C" void kernel_launch(void* const* d_in, const int* in_sizes, int n_in,
                              void* d_out, int out_size, void* d_ws, size_t ws_size,
                              hipStream_t stream) {
    const float* node_feats = (const float*)d_in[0];
    const float* edge_attrs = (const float*)d_in[1];
    const int*   senders    = (const int*)d_in[2];
    const int*   receivers  = (const int*)d_in[3];
    const float* w0         = (const float*)d_in[4];
    const float* w1         = (const float*)d_in[5];
    const float* w2         = (const float*)d_in[6];
    float* out = (float*)d_out;

    int n_edges = in_sizes[1] / 4;
    long long n_out = (long long)out_size;

    // Scatter target must start at zero every call (harness poisons d_out).
    int zb = (int)((n_out + 256LL * 8 - 1) / (256LL * 8));
    zero_out_kernel<<<zb, 256, 0, stream>>>(out, n_out);

    int nblocks = (n_edges + EDGES_PER_BLOCK - 1) / EDGES_PER_BLOCK;
    mpconv_kernel<<<nblocks, 256, 0, stream>>>(node_feats, edge_attrs, senders,
                                               receivers, w0, w1, w2, out, n_edges);
}